// Ylm_85813446574544
// MI455X (gfx1250) — compile-verified
//
#include <hip/hip_runtime.h>
#include <math.h>

#define LMAX 10
#define NL (LMAX + 1)
#define NPAIR (NL * NL)

// ---------------------------------------------------------------------------
// Pass 0: zero the global "flag" word in workspace (harness poisons ws).
// ---------------------------------------------------------------------------
__global__ void ylm_zero_flag(unsigned int* __restrict__ flag) {
    if (threadIdx.x == 0) flag[0] = 0u;
}

// ---------------------------------------------------------------------------
// Pass 1: global any() reduction:  flag |= (|x| < 0.01|z|) && (|y| < 0.01|z|)
// Predicate is rare for N(0,1) inputs, so a guarded atomicOr is cheap.
// ---------------------------------------------------------------------------
__global__ __launch_bounds__(256) void ylm_flag_kernel(
    const float* __restrict__ xyz, int n, unsigned int* __restrict__ flag) {
    int i = blockIdx.x * blockDim.x + threadIdx.x;
    if (i >= n) return;
    float x = xyz[3 * i + 0];
    float y = xyz[3 * i + 1];
    float z = xyz[3 * i + 2];
    float tol = 0.01f * fabsf(z);
    if ((fabsf(x) < tol) && (fabsf(y) < tol)) {
        atomicOr(flag, 1u);
    }
}

// ---------------------------------------------------------------------------
// Pass 2: main Ylm + gradient kernel. One lane per point; all per-point state
// (66-entry ALP triangle + 11+11 trig values) lives in VGPRs via full unroll.
// Store-bandwidth bound (484 floats/point out) -> non-temporal stores.
// ---------------------------------------------------------------------------
__global__ __launch_bounds__(256) void ylm_main_kernel(
    const float* __restrict__ xyz, const unsigned int* __restrict__ flagp,
    float* __restrict__ out, int n) {
    const int tid = blockIdx.x * blockDim.x + threadIdx.x;
    if (tid >= n) return;

    // Uniform scalar load of the global flag -> uniform tilt angle.
    const float angle = (flagp[0] != 0u) ? 0.01f : 0.0f;

    const float x0 = xyz[3 * tid + 0];
    const float y0 = xyz[3 * tid + 1];
    const float z0 = xyz[3 * tid + 2];

    const float x = x0;
    const float y = fmaf(-angle, z0, y0);
    const float z = fmaf(angle, y0, z0);

    const float rxy_sq = fmaf(x, x, y * y);
    const float rxy    = sqrtf(rxy_sq);
    const float r      = sqrtf(fmaf(z, z, rxy_sq));

    const float inv_r   = 1.0f / r;
    const float inv_rxy = 1.0f / rxy;

    const float st = rxy * inv_r;   // sin(theta)
    const float ct = z * inv_r;     // cos(theta)
    const float sp = y * inv_rxy;   // sin(phi)
    const float cp = x * inv_rxy;   // cos(phi)
    const float inv_st = r * inv_rxy;  // 1/sin(theta)

    const float r2   = r * r;
    const float r_ct = r * ct;
    const float r_st = r * st;

    // --- Associated Legendre triangle (scaled by r^l), alp[l*(l+1)/2 + m] ---
    float alp[(NL * (NL + 1)) / 2];
    alp[0] = 0.28209479177387814f;  // sqrt(1/4pi)

#pragma unroll
    for (int l = 1; l <= LMAX; ++l) {
        const int TL  = (l * (l + 1)) / 2;
        const int TL1 = ((l - 1) * l) / 2;
        const int TL2 = ((l - 2) * (l - 1)) / 2;
#pragma unroll
        for (int m = 0; m <= l - 2; ++m) {
            const float a = (float)sqrt((4.0 * l * l - 1.0) / (double)(l * l - m * m));
            const float b = -(float)sqrt((double)((l - 1) * (l - 1) - m * m) /
                                         (4.0 * (l - 1) * (l - 1) - 1.0));
            alp[TL + m] = a * fmaf(r2 * b, alp[TL2 + m], r_ct * alp[TL1 + m]);
        }
        const float c = (float)sqrt(2.0 * l + 1.0);
        const float d = -(float)sqrt(1.0 + 0.5 / (double)l);
        alp[TL + l - 1] = c * r_ct * alp[TL1 + l - 1];
        alp[TL + l]     = d * r_st * alp[TL1 + l - 1];
    }

    // --- sin(m*phi), cos(m*phi) via Chebyshev recurrence ---
    float sins[NL], coss[NL];
    sins[0] = 0.0f; coss[0] = 1.0f;
    sins[1] = sp;   coss[1] = cp;
#pragma unroll
    for (int m = 2; m <= LMAX; ++m) {
        sins[m] = fmaf(sp, coss[m - 1], cp * sins[m - 1]);
        coss[m] = fmaf(cp, coss[m - 1], -(sp * sins[m - 1]));
    }

    // Hoisted angular products for the Cartesian rotation.
    const float stcp = st * cp, stsp = st * sp;
    const float ctcp = ct * cp, ctsp = ct * sp;

    float* __restrict__ Yout  = out;                        // [11][11][N]
    float* __restrict__ dYout = out + (size_t)NPAIR * n;    // [11][11][N][3]

#pragma unroll
    for (int i = 0; i <= LMAX; ++i) {
#pragma unroll
        for (int j = 0; j <= LMAX; ++j) {
            const int   L  = (i > j) ? i : j;
            const int   M  = (i > j) ? (i - j) : (j - i);
            const float A  = alp[(L * (L + 1)) / 2 + M];
            const float tl = (i >= j) ? coss[M] : sins[M];  // trig for Y[i][j]
            const float tu = (j >= i) ? coss[M] : sins[M];  // trig for Y[j][i]

            const float Yij = A * tl;
            const float Lf  = (float)L;
            const float Y_r = Lf * Yij * inv_r;

            float Y_theta = ct * Lf * Yij * inv_st;
            if (i >= 1 && j >= 1) {
                // Y[i-1][j-1]: same M, L-1, same upper/lower trig selection.
                const float Yp   = alp[((L - 1) * L) / 2 + M] * tl;
                const float coef = (float)sqrt(
                    (double)((L - M) * (L + M) * (2 * L + 1)) / (double)(2 * L - 1));
                Y_theta = fmaf(-(r * coef * inv_st), Yp, Y_theta);
            }

            const float sgn     = (i > j) ? -1.0f : 1.0f;
            const float Y_phi   = A * tu * (sgn * (float)M);
            const float F_theta = Y_theta * inv_r;
            const float F_phi   = Y_phi * (inv_r * inv_st);

            const float F_x = fmaf(stcp, Y_r, fmaf(ctcp, F_theta, -(sp * F_phi)));
            const float F_y = fmaf(stsp, Y_r, fmaf(ctsp, F_theta,  cp * F_phi));
            const float F_z = fmaf(ct,   Y_r, -(st * F_theta));

            const int    p    = i * NL + j;
            const size_t yidx = (size_t)p * (size_t)n + (size_t)tid;

            __builtin_nontemporal_store(Yij, Yout + yidx);

            const size_t db = yidx * 3u;
            __builtin_nontemporal_store(F_x,                     dYout + db + 0);
            __builtin_nontemporal_store(fmaf(angle, F_z, F_y),   dYout + db + 1);
            __builtin_nontemporal_store(fmaf(-angle, F_y, F_z),  dYout + db + 2);
        }
    }
}

// ---------------------------------------------------------------------------
extern "C" void kernel_launch(void* const* d_in, const int* in_sizes, int n_in,
                              void* d_out, int out_size, void* d_ws, size_t ws_size,
                              hipStream_t stream) {
    const float* xyz = (const float*)d_in[0];
    const int n = in_sizes[0] / 3;  // xyz is (N, 3) float32
    float* out = (float*)d_out;
    unsigned int* flag = (unsigned int*)d_ws;

    const int block = 256;
    const int grid = (n + block - 1) / block;

    hipLaunchKernelGGL(ylm_zero_flag, dim3(1), dim3(1), 0, stream, flag);
    hipLaunchKernelGGL(ylm_flag_kernel, dim3(grid), dim3(block), 0, stream, xyz, n, flag);
    hipLaunchKernelGGL(ylm_main_kernel, dim3(grid), dim3(block), 0, stream, xyz, flag, out, n);
}